// RNNModel_47785806135834
// MI455X (gfx1250) — compile-verified
//
#include <hip/hip_runtime.h>
#include <hip/hip_bf16.h>
#include <math.h>

// ---------------------------------------------------------------------------
// tanh-RNN for MI455X (gfx1250, wave32, WMMA).
//   T=512, B=256, I=256, H=512
// Phase 1: xp = x @ w_ih^T + b_ih + b_hh           (GEMM, full chip)
// Phase 2: h_t = tanh(xp_t + h_{t-1} @ w_hh^T)     (persistent, LDS-resident h)
// Phase 3: out = hs @ w_fc^T + b_fc                (GEMM, full chip)
// All matrix math on V_WMMA_F32_16X16X4_F32 (native fp32 matrix pipe).
// GEMM waves are register-blocked 2Mx4N (8 WMMA per 6 b64 loads).
// Recurrence waves are register-blocked 2Mx1N sharing every w_hh load.
// ---------------------------------------------------------------------------

typedef float v2f __attribute__((ext_vector_type(2)));
typedef float v8f __attribute__((ext_vector_type(8)));

#define TT 512
#define BB 256
#define II 256
#define HH 512
#define HP (HH + 4)   // LDS row pitch (floats): 516 % 64 = 4 -> conflict-free
#define BTILE 32      // batch rows per recurrence workgroup

// fp32 WMMA A layout (ISA 7.12.2): lane<16 holds M=lane, K=k0..k0+1 in 2 VGPRs;
// lane>=16 holds M=lane-16, K=k0+2..k0+3. B mirrors with N in place of M.
// C/D layout: VGPR r = row (lane>=16 ? 8:0)+r, col = lane&15.

// OUT[m0:m0+32, n0:n0+64] = A[m0:,:K] * W[n0:,:K]^T + bias (+bias2)
__global__ __launch_bounds__(128) void
gemm_xwT_bias(const float* __restrict__ A,     // [M,K]
              const float* __restrict__ W,     // [N,K]
              const float* __restrict__ bias,  // [N]
              const float* __restrict__ bias2, // [N] or nullptr
              float* __restrict__ OUT,         // [M,N]
              int M, int N, int K) {
  const int lane    = threadIdx.x & 31;
  const int wtile   = (blockIdx.x * blockDim.x + threadIdx.x) >> 5;  // uniform
  const int nstrips = N >> 6;                 // 64-wide N strips
  const int m0 = (wtile / nstrips) << 5;      // 32-row M strip
  const int n0 = (wtile % nstrips) << 6;
  if (m0 >= M) return;

  const int row   = lane & 15;
  const int khalf = (lane >> 4) << 1;         // 0 or 2

  const float* __restrict__ arow0 = A + (size_t)(m0 + row) * K;
  const float* __restrict__ arow1 = A + (size_t)(m0 + 16 + row) * K;
  const float* __restrict__ brow0 = W + (size_t)(n0 + row) * K;
  const float* __restrict__ brow1 = W + (size_t)(n0 + 16 + row) * K;
  const float* __restrict__ brow2 = W + (size_t)(n0 + 32 + row) * K;
  const float* __restrict__ brow3 = W + (size_t)(n0 + 48 + row) * K;

  v8f acc[2][4] = {};
  for (int k0 = 0; k0 < K; k0 += 4) {
    const int kb = k0 + khalf;                       // 8B aligned
    v2f a0 = *(const v2f*)(arow0 + kb);
    v2f a1 = *(const v2f*)(arow1 + kb);
    v2f b0 = *(const v2f*)(brow0 + kb);
    v2f b1 = *(const v2f*)(brow1 + kb);
    v2f b2 = *(const v2f*)(brow2 + kb);
    v2f b3 = *(const v2f*)(brow3 + kb);
    acc[0][0] = __builtin_amdgcn_wmma_f32_16x16x4_f32(false, a0, false, b0, (short)0, acc[0][0], false, false);
    acc[1][0] = __builtin_amdgcn_wmma_f32_16x16x4_f32(false, a1, false, b0, (short)0, acc[1][0], false, false);
    acc[0][1] = __builtin_amdgcn_wmma_f32_16x16x4_f32(false, a0, false, b1, (short)0, acc[0][1], false, false);
    acc[1][1] = __builtin_amdgcn_wmma_f32_16x16x4_f32(false, a1, false, b1, (short)0, acc[1][1], false, false);
    acc[0][2] = __builtin_amdgcn_wmma_f32_16x16x4_f32(false, a0, false, b2, (short)0, acc[0][2], false, false);
    acc[1][2] = __builtin_amdgcn_wmma_f32_16x16x4_f32(false, a1, false, b2, (short)0, acc[1][2], false, false);
    acc[0][3] = __builtin_amdgcn_wmma_f32_16x16x4_f32(false, a0, false, b3, (short)0, acc[0][3], false, false);
    acc[1][3] = __builtin_amdgcn_wmma_f32_16x16x4_f32(false, a1, false, b3, (short)0, acc[1][3], false, false);
  }

  const int col = lane & 15;
  const int mb  = (lane >> 4) << 3;
#pragma unroll
  for (int j = 0; j < 4; ++j) {
    const int n = n0 + (j << 4) + col;
    const float bs = bias[n] + (bias2 ? bias2[n] : 0.0f);
#pragma unroll
    for (int mt = 0; mt < 2; ++mt) {
      const int mbase = m0 + (mt << 4) + mb;
#pragma unroll
      for (int r = 0; r < 8; ++r)
        OUT[(size_t)(mbase + r) * N + n] = acc[mt][j][r] + bs;
    }
  }
}

// Persistent recurrence: 8 workgroups, each owns 32 batch rows for all T steps.
// 1024 threads = 32 waves; wave w owns hidden column tile n0 = 16*w and both
// 16-row batch halves (two accumulators share every w_hh load).
// h double-buffered in dynamic LDS (2 * 32 * 516 * 4B = 132 KB of 320 KB).
// XP is read as xp[t] and overwritten in place with h_t (becomes hs).
__global__ __launch_bounds__(1024) void
rnn_recurrence(const float* __restrict__ Whh,  // [H,H] row-major
               float* __restrict__ XP) {       // [T,B,H] : xp in, hs out
  extern __shared__ float smem[];
  const int BUF = BTILE * HP;

  const int tid = threadIdx.x;
  for (int i = tid; i < 2 * BUF; i += 1024) smem[i] = 0.0f;
  __syncthreads();

  const int lane  = tid & 31;
  const int wave  = tid >> 5;
  const int n0    = wave << 4;              // hidden tile
  const int bb0   = blockIdx.x * BTILE;     // batch slice
  const int row   = lane & 15;
  const int khalf = (lane >> 4) << 1;
  const int col   = lane & 15;
  const int mb    = (lane >> 4) << 3;

  const float* __restrict__ brow = Whh + (size_t)(n0 + col) * HH;

  for (int t = 0; t < TT; ++t) {
    const int roff = (t & 1) * BUF;
    const int woff = BUF - roff;
    const float* hr = smem + roff;
    float* hw       = smem + woff;

    v8f acc0 = {}, acc1 = {};
#pragma unroll 4
    for (int k0 = 0; k0 < HH; k0 += 4) {
      const int kb = k0 + khalf;
      v2f b  = *(const v2f*)(brow + kb);                  // w_hh[n][kb..] (L2)
      v2f a0 = *(const v2f*)(hr + row * HP + kb);         // h rows 0..15 (LDS)
      v2f a1 = *(const v2f*)(hr + (16 + row) * HP + kb);  // h rows 16..31
      acc0 = __builtin_amdgcn_wmma_f32_16x16x4_f32(false, a0, false, b, (short)0, acc0, false, false);
      acc1 = __builtin_amdgcn_wmma_f32_16x16x4_f32(false, a1, false, b, (short)0, acc1, false, false);
    }

    float* __restrict__ xpt = XP + ((size_t)t * BB + bb0) * HH;
#pragma unroll
    for (int r = 0; r < 8; ++r) {
      const size_t g0 = (size_t)(mb + r) * HH + (n0 + col);
      const size_t g1 = (size_t)(16 + mb + r) * HH + (n0 + col);
      float v0 = tanhf(acc0[r] + xpt[g0]);
      float v1 = tanhf(acc1[r] + xpt[g1]);
      hw[(mb + r) * HP + (n0 + col)]      = v0;  // next-step state
      hw[(16 + mb + r) * HP + (n0 + col)] = v1;
      xpt[g0] = v0;                              // hs[t] in place of xp[t]
      xpt[g1] = v1;
    }
    // Prefetch next step's xp tile (HBM) behind the barrier + next K-loop.
    if (t + 1 < TT) {
      const float* nxt = XP + ((size_t)(t + 1) * BB + bb0 + lane) * HH + n0;
      __builtin_prefetch(nxt, 0, 1);
    }
    __syncthreads();  // hw complete before next step reads it
  }
}

extern "C" void kernel_launch(void* const* d_in, const int* in_sizes, int n_in,
                              void* d_out, int out_size, void* d_ws, size_t ws_size,
                              hipStream_t stream) {
  const float* x    = (const float*)d_in[0];  // [T,B,I]
  const float* w_ih = (const float*)d_in[1];  // [H,I]
  const float* w_hh = (const float*)d_in[2];  // [H,H]
  const float* b_ih = (const float*)d_in[3];  // [H]
  const float* b_hh = (const float*)d_in[4];  // [H]
  const float* w_fc = (const float*)d_in[5];  // [I,H]
  const float* b_fc = (const float*)d_in[6];  // [I]
  float* out = (float*)d_out;                 // [T,B,I]
  float* xp  = (float*)d_ws;                  // [T,B,H] scratch (256 MB)

  const int M = TT * BB;  // 131072

  // Phase 1: xp = x @ w_ih^T + (b_ih + b_hh).  (M/32)x(H/64) wave strips.
  {
    int waves  = (M / 32) * (HH / 64);  // 32768
    int blocks = waves / 4;             // 4 waves / 128-thread block
    gemm_xwT_bias<<<blocks, 128, 0, stream>>>(x, w_ih, b_ih, b_hh, xp, M, HH, II);
  }

  // Phase 2: persistent recurrence, 8 workgroups (one per 32-batch slice).
  {
    size_t lds = (size_t)2 * BTILE * HP * sizeof(float);  // 132 KB
    rnn_recurrence<<<BB / BTILE, 1024, lds, stream>>>(w_hh, xp);
  }

  // Phase 3: out = hs @ w_fc^T + b_fc.  (M/32)x(I/64) wave strips.
  {
    int waves  = (M / 32) * (II / 64);  // 16384
    int blocks = waves / 4;
    gemm_xwT_bias<<<blocks, 128, 0, stream>>>(xp, w_fc, b_fc, nullptr, out, M, II, HH);
  }
}